// UMLPatternEmbedding_59846074303063
// MI455X (gfx1250) — compile-verified
//
#include <hip/hip_runtime.h>

typedef __attribute__((ext_vector_type(2))) float v2f;
typedef __attribute__((ext_vector_type(8))) float v8f;

#define NODES_PER_TILE 16
#define WAVES_PER_BLOCK 4

// ---------------------------------------------------------------------------
// WMMA fp32 GEMM: H[M x N] = X[M x K] @ W[K x N], row-major, N = 16*NT.
// One wave -> 16 rows x full N. W staged in LDS. M must be multiple of 16.
// ---------------------------------------------------------------------------
template <int NT>
__global__ void gemm_wmma_f32(const float* __restrict__ X,
                              const float* __restrict__ W,
                              float* __restrict__ H,
                              int Mtiles, int K) {
    const int N = NT * 16;
    extern __shared__ float lW[];  // K * N floats

    // cooperative load of W into LDS
    for (int i = threadIdx.x; i < K * N; i += blockDim.x) lW[i] = W[i];
    __syncthreads();

    const int lane = threadIdx.x & 31;
    const int wave = threadIdx.x >> 5;
    const int tile = blockIdx.x * WAVES_PER_BLOCK + wave;
    if (tile >= Mtiles) return;   // wave-uniform; EXEC stays all-ones for WMMA

    const int m15 = lane & 15;    // row within tile (A), col within tile (B/C)
    const int hi  = lane >> 4;    // 0: K={0,1}, 1: K={2,3}
    const int row = tile * 16 + m15;
    const float* xrow = X + (size_t)row * K;

    v8f acc[NT];
#pragma unroll
    for (int t = 0; t < NT; ++t) acc[t] = (v8f){0.f,0.f,0.f,0.f,0.f,0.f,0.f,0.f};

    for (int k0 = 0; k0 < K; k0 += 4) {
        // A fragment: contiguous float2 at K-offset k0 + 2*hi
        const float2 af = *(const float2*)(xrow + k0 + 2 * hi);
        v2f a; a.x = af.x; a.y = af.y;
#pragma unroll
        for (int t = 0; t < NT; ++t) {
            const int col = t * 16 + m15;
            v2f b;
            b.x = lW[(k0 + 2 * hi)     * N + col];
            b.y = lW[(k0 + 2 * hi + 1) * N + col];
            acc[t] = __builtin_amdgcn_wmma_f32_16x16x4_f32(
                false, a, false, b, (short)0, acc[t], false, false);
        }
    }

    // C/D layout: VGPR r -> M = r + 8*hi, N = lane&15
    const int mbase = tile * 16 + 8 * hi;
#pragma unroll
    for (int t = 0; t < NT; ++t) {
        const int col = t * 16 + m15;
#pragma unroll
        for (int r = 0; r < 8; ++r) {
            H[(size_t)(mbase + r) * N + col] = acc[t][r];
        }
    }
}

// ---------------------------------------------------------------------------
// helpers
// ---------------------------------------------------------------------------
__global__ void fill_kernel(float* __restrict__ p, float v, long long n) {
    long long i = (long long)blockIdx.x * blockDim.x + threadIdx.x;
    if (i < n) p[i] = v;
}

__global__ void edge_degree_kernel(const int* __restrict__ dst,
                                   float* __restrict__ deg, long long E) {
    long long e = (long long)blockIdx.x * blockDim.x + threadIdx.x;
    if (e < E) atomicAdd(&deg[dst[e]], 1.0f);
}

__global__ void rsqrt_kernel(float* __restrict__ d, long long n) {
    long long i = (long long)blockIdx.x * blockDim.x + threadIdx.x;
    if (i < n) d[i] = rsqrtf(fmaxf(d[i], 1.0f));
}

// A[dst] += H[src] * dinv[src]*dinv[dst], 4 features per thread (float4 gather)
__global__ void edge_aggregate_kernel(const int* __restrict__ src,
                                      const int* __restrict__ dst,
                                      const float* __restrict__ dinv,
                                      const float* __restrict__ H,
                                      float* __restrict__ A,
                                      long long E, int F) {
    const int FQ = F >> 2;
    long long idx = (long long)blockIdx.x * blockDim.x + threadIdx.x;
    long long e = idx / FQ;
    if (e >= E) return;
    int q = (int)(idx - e * FQ);
    int s = src[e], d = dst[e];
    float w = dinv[s] * dinv[d];
    const float4 h = *(const float4*)(H + (size_t)s * F + 4 * q);
    float* ap = A + (size_t)d * F + 4 * q;
    atomicAdd(ap + 0, h.x * w);
    atomicAdd(ap + 1, h.y * w);
    atomicAdd(ap + 2, h.z * w);
    atomicAdd(ap + 3, h.w * w);
}

// A = [relu]( A + H*dinv^2 + b )   (self-loop folded in, no atomics)
__global__ void epilogue_kernel(float* __restrict__ A,
                                const float* __restrict__ H,
                                const float* __restrict__ dinv,
                                const float* __restrict__ bias,
                                long long n, int F, int doRelu) {
    long long idx = (long long)blockIdx.x * blockDim.x + threadIdx.x;
    if (idx >= n * F) return;
    long long i = idx / F;
    int f = (int)(idx - i * F);
    float di = dinv[i];
    float v = A[idx] + H[idx] * di * di + bias[f];
    if (doRelu) v = fmaxf(v, 0.0f);
    A[idx] = v;
}

__global__ void pool_count_kernel(const int* __restrict__ batch,
                                  float* __restrict__ cnts, long long n) {
    long long i = (long long)blockIdx.x * blockDim.x + threadIdx.x;
    if (i < n) atomicAdd(&cnts[batch[i]], 1.0f);
}

__global__ void pool_sum_kernel(const int* __restrict__ batch,
                                const float* __restrict__ A,
                                float* __restrict__ sums,
                                long long n, int F) {
    long long idx = (long long)blockIdx.x * blockDim.x + threadIdx.x;
    if (idx >= n * F) return;
    long long i = idx / F;
    int f = (int)(idx - i * F);
    atomicAdd(&sums[(size_t)batch[i] * F + f], A[idx]);
}

__global__ void pool_div_kernel(const float* __restrict__ sums,
                                const float* __restrict__ cnts,
                                float* __restrict__ out, int G, int F) {
    int idx = blockIdx.x * blockDim.x + threadIdx.x;
    if (idx >= G * F) return;
    out[idx] = sums[idx] / fmaxf(cnts[idx / F], 1.0f);
}

// ---------------------------------------------------------------------------
static inline int cdiv_ll(long long a, long long b) { return (int)((a + b - 1) / b); }

extern "C" void kernel_launch(void* const* d_in, const int* in_sizes, int n_in,
                              void* d_out, int out_size, void* d_ws, size_t ws_size,
                              hipStream_t stream) {
    const float* x     = (const float*)d_in[0];
    const int*   eidx  = (const int*)  d_in[1];
    const int*   batch = (const int*)  d_in[2];
    const float* W1 = (const float*)d_in[3]; const float* b1 = (const float*)d_in[4];
    const float* W2 = (const float*)d_in[5]; const float* b2 = (const float*)d_in[6];
    const float* W3 = (const float*)d_in[7]; const float* b3 = (const float*)d_in[8];
    float* out = (float*)d_out;

    const long long Nn  = in_sizes[2];          // 100000
    const long long E   = in_sizes[1] / 2;      // 3200000
    const int IN  = (int)(in_sizes[0] / Nn);    // 128
    const int HID = in_sizes[4];                // 64
    const int EMB = in_sizes[8];                // 32
    const int G   = 256;

    const int* src = eidx;
    const int* dst = eidx + E;

    // workspace layout (floats)
    float* ws   = (float*)d_ws;
    float* Hbuf = ws;                    // Nn * HID
    float* Abuf = Hbuf + Nn * HID;       // Nn * HID
    float* dinv = Abuf + Nn * HID;       // Nn   (holds deg first)
    float* sums = dinv + Nn;             // G * EMB
    float* cnts = sums + (size_t)G * EMB;// G

    const int TPB = 256;
    const int Mtiles = (int)((Nn + 15) / 16);
    const int gemmBlocks = (Mtiles + WAVES_PER_BLOCK - 1) / WAVES_PER_BLOCK;
    const int gemmThreads = 32 * WAVES_PER_BLOCK;

    // ---- normalization: deg (with self loop) -> dinv ----
    fill_kernel<<<cdiv_ll(Nn, TPB), TPB, 0, stream>>>(dinv, 1.0f, Nn);
    edge_degree_kernel<<<cdiv_ll(E, TPB), TPB, 0, stream>>>(dst, dinv, E);
    rsqrt_kernel<<<cdiv_ll(Nn, TPB), TPB, 0, stream>>>(dinv, Nn);

    // ---- layer 1: 128 -> 64, relu ----
    gemm_wmma_f32<4><<<gemmBlocks, gemmThreads, IN * HID * sizeof(float), stream>>>(
        x, W1, Hbuf, Mtiles, IN);
    fill_kernel<<<cdiv_ll(Nn * HID, TPB), TPB, 0, stream>>>(Abuf, 0.0f, Nn * HID);
    edge_aggregate_kernel<<<cdiv_ll(E * (HID / 4), TPB), TPB, 0, stream>>>(
        src, dst, dinv, Hbuf, Abuf, E, HID);
    epilogue_kernel<<<cdiv_ll(Nn * HID, TPB), TPB, 0, stream>>>(
        Abuf, Hbuf, dinv, b1, Nn, HID, 1);

    // ---- layer 2: 64 -> 64, relu ----
    gemm_wmma_f32<4><<<gemmBlocks, gemmThreads, HID * HID * sizeof(float), stream>>>(
        Abuf, W2, Hbuf, Mtiles, HID);
    fill_kernel<<<cdiv_ll(Nn * HID, TPB), TPB, 0, stream>>>(Abuf, 0.0f, Nn * HID);
    edge_aggregate_kernel<<<cdiv_ll(E * (HID / 4), TPB), TPB, 0, stream>>>(
        src, dst, dinv, Hbuf, Abuf, E, HID);
    epilogue_kernel<<<cdiv_ll(Nn * HID, TPB), TPB, 0, stream>>>(
        Abuf, Hbuf, dinv, b2, Nn, HID, 1);

    // ---- layer 3: 64 -> 32, no relu ----
    gemm_wmma_f32<2><<<gemmBlocks, gemmThreads, HID * EMB * sizeof(float), stream>>>(
        Abuf, W3, Hbuf, Mtiles, HID);
    fill_kernel<<<cdiv_ll(Nn * EMB, TPB), TPB, 0, stream>>>(Abuf, 0.0f, Nn * EMB);
    edge_aggregate_kernel<<<cdiv_ll(E * (EMB / 4), TPB), TPB, 0, stream>>>(
        src, dst, dinv, Hbuf, Abuf, E, EMB);
    epilogue_kernel<<<cdiv_ll(Nn * EMB, TPB), TPB, 0, stream>>>(
        Abuf, Hbuf, dinv, b3, Nn, EMB, 0);

    // ---- global mean pool ----
    fill_kernel<<<cdiv_ll((long long)G * EMB + G, TPB), TPB, 0, stream>>>(
        sums, 0.0f, (long long)G * EMB + G);  // zero sums and cnts (contiguous)
    pool_count_kernel<<<cdiv_ll(Nn, TPB), TPB, 0, stream>>>(batch, cnts, Nn);
    pool_sum_kernel<<<cdiv_ll(Nn * EMB, TPB), TPB, 0, stream>>>(batch, Abuf, sums, Nn, EMB);
    pool_div_kernel<<<cdiv_ll((long long)G * EMB, TPB), TPB, 0, stream>>>(
        sums, cnts, out, G, EMB);
}